// TransitionDown_23287312679062
// MI455X (gfx1250) — compile-verified
//
#include <hip/hip_runtime.h>
#include <hip/hip_bf16.h>

// -------- problem constants (from reference) --------
#define NPTS   16384
#define CIN    64
#define COUT   128
#define NSAMP  16
#define MTOT   8192          // N / (STRIDE/2)
#define M1     4096
#define KDIM   67            // 3 + CIN
#define KPAD   68            // padded to multiple of 4 for 16x16x4 f32 WMMA
#define BN_EPS 1e-5f

// -------- d_out layout (floats, concatenated in return order) --------
#define OFF_NP   0                         // n_p   [4096,3]
#define OFF_X1   12288                     // x1    [4096,128]
#define OFF_NO   536576                    // n_o   [1]
#define OFF_NP2  536577                    // n_p2  [4096,3]
#define OFF_X2   548865                    // x2    [4096,128]
#define OFF_NO2  1073153                   // n_o2  [1]

// -------- workspace layout (bytes) --------
#define WS_FPSIDX  0                       // 8192 * 4
#define WS_KNNIDX  32768                   // 8192*16*4
#define WS_STATS   557056                  // 512 floats: sum[2][128], sumsq[2][128]
#define WS_HBUF    559104                  // 8192*16*128*4 = 64 MB

typedef __attribute__((ext_vector_type(2))) float v2f;
typedef __attribute__((ext_vector_type(8))) float v8f;

// ============================================================
// init: zero BN stats, write n_o / n_o2
// ============================================================
__global__ void init_kernel(float* stats, float* out) {
    int t = threadIdx.x;
    if (t < 512) stats[t] = 0.0f;
    if (t == 0) out[OFF_NO]  = 4096.0f;
    if (t == 1) out[OFF_NO2] = 4096.0f;
}

// ============================================================
// FPS: single persistent workgroup (1024 thr = 32 waves on one WGP).
// Each thread owns 16 points (coords + running min-dist in registers).
// Per iteration: dist update, wave32 shuffle argmax, LDS cross-wave argmax.
// ============================================================
__global__ __launch_bounds__(1024) void fps_kernel(const float* __restrict__ p,
                                                   int* __restrict__ fpsIdx) {
    const int tid  = threadIdx.x;
    const int lane = tid & 31;
    const int wid  = tid >> 5;

    __shared__ float sVal[32];
    __shared__ int   sIdx[32];
    __shared__ int   sBest;

    float lx[16], ly[16], lz[16], ld[16];
#pragma unroll
    for (int k = 0; k < 16; ++k) {
        int gi = k * 1024 + tid;
        lx[k] = p[gi * 3 + 0];
        ly[k] = p[gi * 3 + 1];
        lz[k] = p[gi * 3 + 2];
        ld[k] = 1e10f;
    }
    if (tid == 0) fpsIdx[0] = 0;
    int last = 0;

    for (int i = 1; i < MTOT; ++i) {
        // broadcast load of last chosen point (uniform address)
        float px = p[last * 3 + 0];
        float py = p[last * 3 + 1];
        float pz = p[last * 3 + 2];

        float bv = -1.0f;
        int   bi = 0x7fffffff;
#pragma unroll
        for (int k = 0; k < 16; ++k) {
            float dx = lx[k] - px, dy = ly[k] - py, dz = lz[k] - pz;
            float d  = fmaf(dx, dx, fmaf(dy, dy, dz * dz));
            d = fminf(ld[k], d);
            ld[k] = d;
            int gi = k * 1024 + tid;          // ascending -> strict > keeps earliest
            if (d > bv) { bv = d; bi = gi; }
        }
        // wave32 argmax (tie: lowest index, matching jnp.argmax)
#pragma unroll
        for (int off = 16; off > 0; off >>= 1) {
            float ov = __shfl_xor(bv, off, 32);
            int   oi = __shfl_xor(bi, off, 32);
            if (ov > bv || (ov == bv && oi < bi)) { bv = ov; bi = oi; }
        }
        if (lane == 0) { sVal[wid] = bv; sIdx[wid] = bi; }
        __syncthreads();
        if (wid == 0) {
            float v2 = sVal[lane];
            int   i2 = sIdx[lane];
#pragma unroll
            for (int off = 16; off > 0; off >>= 1) {
                float ov = __shfl_xor(v2, off, 32);
                int   oi = __shfl_xor(i2, off, 32);
                if (ov > v2 || (ov == v2 && oi < i2)) { v2 = ov; i2 = oi; }
            }
            if (lane == 0) { sBest = i2; fpsIdx[i] = i2; }
        }
        __syncthreads();
        last = sBest;
    }
}

// ============================================================
// KNN: one query per thread; brute-force scan of 16384 points
// (uniform broadcast loads, L2-resident). Register-resident sorted
// top-16, rare-branch insertion. Also writes n_p / n_p2 coords.
// ============================================================
__global__ __launch_bounds__(256) void knn_kernel(const float* __restrict__ p,
                                                  const int* __restrict__ fpsIdx,
                                                  int* __restrict__ knnIdx,
                                                  float* __restrict__ out) {
    int q = blockIdx.x * 256 + threadIdx.x;
    if (q >= MTOT) return;

    int qi = fpsIdx[q];
    float qx = p[qi * 3 + 0], qy = p[qi * 3 + 1], qz = p[qi * 3 + 2];

    float* np = (q < M1) ? (out + OFF_NP + q * 3)
                         : (out + OFF_NP2 + (q - M1) * 3);
    np[0] = qx; np[1] = qy; np[2] = qz;

    float bd[NSAMP];
    int   bi[NSAMP];
#pragma unroll
    for (int j = 0; j < NSAMP; ++j) { bd[j] = 3.4e38f; bi[j] = 0; }

    for (int k = 0; k < NPTS; ++k) {
        float dx = p[k * 3 + 0] - qx;
        float dy = p[k * 3 + 1] - qy;
        float dz = p[k * 3 + 2] - qz;
        float d  = fmaf(dx, dx, fmaf(dy, dy, dz * dz));
        if (d < bd[NSAMP - 1]) {
            float cd = d; int ci = k;
#pragma unroll
            for (int j = 0; j < NSAMP; ++j) {
                if (cd < bd[j]) {
                    float td = bd[j]; int ti = bi[j];
                    bd[j] = cd; bi[j] = ci;
                    cd = td; ci = ti;
                }
            }
        }
    }
#pragma unroll
    for (int j = 0; j < NSAMP; ++j) knnIdx[q * NSAMP + j] = bi[j];
}

// ============================================================
// Gather + GEMM via V_WMMA_F32_16X16X4_F32 (full f32 precision).
// Block = 128 thr (4 waves); each wave computes one query's
// h[16,128] = feats[16,68(pad)] @ W[68(pad),128]:
//   8 output tiles of 16x16, 17 K-steps of 4.
// Also accumulates per-channel BN sum/sumsq (LDS ds_add_f32 ->
// one global atomicAdd per channel per block; blocks never
// straddle the m1/m2 half boundary).
// ============================================================
__global__ __launch_bounds__(128) void gemm_kernel(const float* __restrict__ p,
                                                   const float* __restrict__ x,
                                                   const float* __restrict__ W,
                                                   const int* __restrict__ fpsIdx,
                                                   const int* __restrict__ knnIdx,
                                                   float* __restrict__ hbuf,
                                                   float* __restrict__ stats) {
    __shared__ float sW[KPAD * COUT];          // 34816 B
    __shared__ float sF[4 * NSAMP * KPAD];     // 17408 B
    __shared__ float sSum[COUT];
    __shared__ float sSqs[COUT];

    const int tid  = threadIdx.x;
    const int lane = tid & 31;
    const int wid  = tid >> 5;                 // 0..3
    const int q    = blockIdx.x * 4 + wid;     // 0..8191 (2048 blocks exactly)
    const int half = (blockIdx.x < 1024) ? 0 : 1;

    // W -> LDS (padded row 67 zeroed)
    for (int i = tid; i < KPAD * COUT; i += 128) {
        int k = i >> 7, n = i & (COUT - 1);
        sW[i] = (k < KDIM) ? W[k * COUT + n] : 0.0f;
    }
    if (tid < COUT) { sSum[tid] = 0.0f; sSqs[tid] = 0.0f; }

    // Gather features for this wave's query: rel xyz ++ x[idx], pad col 67 = 0
    {
        const int r  = lane & 15;              // sample row
        const int ch = lane >> 4;              // column half: 0 -> cols 0..33, 1 -> 34..67
        const int qi = fpsIdx[q];
        const float qx = p[qi * 3 + 0], qy = p[qi * 3 + 1], qz = p[qi * 3 + 2];
        const int nb = knnIdx[q * NSAMP + r];
        float* fRow = &sF[(wid * NSAMP + r) * KPAD];
#pragma unroll 2
        for (int j = 0; j < 34; ++j) {
            int col = ch * 34 + j;
            float v;
            if (col == 0)      v = p[nb * 3 + 0] - qx;
            else if (col == 1) v = p[nb * 3 + 1] - qy;
            else if (col == 2) v = p[nb * 3 + 2] - qz;
            else if (col < KDIM) v = x[nb * CIN + (col - 3)];
            else               v = 0.0f;
            fRow[col] = v;
        }
    }
    __syncthreads();

    // ---- WMMA: f32 16x16x4, K = 68 in 17 steps, 8 N-tiles ----
    v8f acc[8] = {};
    const float* fBase = &sF[wid * NSAMP * KPAD];
    const int rr = lane & 15;
    const int kh = (lane >> 4) << 1;           // 0 for lanes 0-15, 2 for 16-31

#pragma unroll 1
    for (int k0 = 0; k0 < KPAD; k0 += 4) {
        const int kk = k0 + kh;
        // A fragment (16x4 f32): VGPR0 = K=k0 / k0+2 across lane halves, VGPR1 = +1
        v2f a;
        a.x = fBase[rr * KPAD + kk];
        a.y = fBase[rr * KPAD + kk + 1];
#pragma unroll
        for (int nt = 0; nt < 8; ++nt) {
            const int n = nt * 16 + rr;
            // B fragment (4x16 f32), mirrored layout
            v2f b;
            b.x = sW[kk * COUT + n];
            b.y = sW[(kk + 1) * COUT + n];
            acc[nt] = __builtin_amdgcn_wmma_f32_16x16x4_f32(
                false, a, false, b, (short)0, acc[nt], false, false);
        }
    }

    // ---- epilogue: store h, accumulate BN sum / sumsq per channel ----
    const int rbase = q * NSAMP + ((lane >> 4) ? 8 : 0);
#pragma unroll
    for (int nt = 0; nt < 8; ++nt) {
        float s = 0.0f, s2 = 0.0f;
#pragma unroll
        for (int v = 0; v < 8; ++v) {
            float e = acc[nt][v];               // row rbase+v, channel nt*16+rr
            hbuf[(rbase + v) * COUT + nt * 16 + rr] = e;
            s  += e;
            s2 += e * e;
        }
        s  += __shfl_xor(s, 16, 32);
        s2 += __shfl_xor(s2, 16, 32);
        if (lane < 16) {
            atomicAdd(&sSum[nt * 16 + rr], s);
            atomicAdd(&sSqs[nt * 16 + rr], s2);
        }
    }
    __syncthreads();
    if (tid < COUT) {
        atomicAdd(&stats[half * COUT + tid], sSum[tid]);
        atomicAdd(&stats[256 + half * COUT + tid], sSqs[tid]);
    }
}

// ============================================================
// BN (train stats per half) + ReLU + max-pool over 16 samples.
// One block per query, one channel per thread (coalesced hbuf reads).
// ============================================================
__global__ __launch_bounds__(128) void bnpool_kernel(const float* __restrict__ hbuf,
                                                     const float* __restrict__ stats,
                                                     const float* __restrict__ gamma,
                                                     const float* __restrict__ beta,
                                                     float* __restrict__ out) {
    const int q = blockIdx.x;
    const int c = threadIdx.x;
    const int half = (q < M1) ? 0 : 1;
    const float inv_cnt = 1.0f / (float)(M1 * NSAMP);

    float mean = stats[half * COUT + c] * inv_cnt;
    float ex2  = stats[256 + half * COUT + c] * inv_cnt;
    float var  = ex2 - mean * mean;
    float scl  = gamma[c] * rsqrtf(var + BN_EPS);
    float sft  = beta[c] - mean * scl;

    float m = 0.0f;                             // relu output is >= 0
#pragma unroll
    for (int r = 0; r < NSAMP; ++r) {
        float v = hbuf[(q * NSAMP + r) * COUT + c];
        v = fmaxf(fmaf(v, scl, sft), 0.0f);
        m = fmaxf(m, v);
    }
    if (q < M1) out[OFF_X1 + q * COUT + c] = m;
    else        out[OFF_X2 + (q - M1) * COUT + c] = m;
}

// ============================================================
extern "C" void kernel_launch(void* const* d_in, const int* in_sizes, int n_in,
                              void* d_out, int out_size, void* d_ws, size_t ws_size,
                              hipStream_t stream) {
    const float* p     = (const float*)d_in[0];
    const float* x     = (const float*)d_in[1];
    // d_in[2] = o (unused; sizes are static)
    const float* W     = (const float*)d_in[3];
    const float* gamma = (const float*)d_in[4];
    const float* beta  = (const float*)d_in[5];
    float* out = (float*)d_out;

    char*  ws      = (char*)d_ws;
    int*   fpsIdx  = (int*)(ws + WS_FPSIDX);
    int*   knnIdx  = (int*)(ws + WS_KNNIDX);
    float* stats   = (float*)(ws + WS_STATS);
    float* hbuf    = (float*)(ws + WS_HBUF);

    init_kernel<<<1, 512, 0, stream>>>(stats, out);
    fps_kernel<<<1, 1024, 0, stream>>>(p, fpsIdx);
    knn_kernel<<<MTOT / 256, 256, 0, stream>>>(p, fpsIdx, knnIdx, out);
    gemm_kernel<<<MTOT / 4, 128, 0, stream>>>(p, x, W, fpsIdx, knnIdx, hbuf, stats);
    bnpool_kernel<<<MTOT, 128, 0, stream>>>(hbuf, stats, gamma, beta, out);
}